// TokenizerEncoder_86096914415818
// MI455X (gfx1250) — compile-verified
//
#include <hip/hip_runtime.h>
#include <hip/hip_bf16.h>

// ---------------------------------------------------------------------------
// TokenizerEncoder chunking for MI455X (gfx1250, wave32).
//  B=8, S=8192, D=128, MAXL=16, MAXC=1024 (hardcoded in reference).
//  d_out = [output_tensor (8*1024*16*128 f32) | output_token_ids (8*1024*16 f32)]
// Pipeline: (1) WMMA f32 16x16x4 logits -> boundary bitmask
//           (2) 8-thread serial recurrence -> compacted chunk ends
//           (3) bandwidth-bound emit of pad(+x) tiles and ids
// ---------------------------------------------------------------------------

#define BATCH 8
#define SEQ   8192
#define DIM   128
#define MAXL  16
#define MAXC  1024

typedef __attribute__((ext_vector_type(2))) float v2f;
typedef __attribute__((ext_vector_type(8))) float v8f;

// --------------------------- Kernel 1: boundaries ---------------------------
// 128 threads (4 waves) per block; block handles 64 consecutive tokens.
// Each wave computes dot(x_t, v) for 16 tokens via 32 x V_WMMA_F32_16X16X4_F32
// (K=128 in steps of 4), with v = W[:,0]-W[:,1] living in column N=0 of B.
// B-fragment masking is done with a per-lane LDS base pointer into a buffer
// whose upper half is zero -> uniform control flow, no EXEC manipulation
// around the WMMA ops (ISA requires EXEC==all-1s for WMMA).
__global__ void tok_boundary_kernel(const float* __restrict__ x,
                                    const float* __restrict__ W,
                                    const float* __restrict__ bias,
                                    unsigned* __restrict__ mask16)
{
    __shared__ float tile[64 * 132];   // 64 tokens x 128 dims, pad to 132 (bank-conflict free)
    __shared__ float vlds[2 * DIM];    // [0:128) = v,  [128:256) = zeros

    const int tid  = threadIdx.x;
    const int tok0 = blockIdx.x * 64;

    // v[d] = W[d][0] - W[d][1]   (W row-major (128,2)); upper half zero.
    vlds[tid]       = W[2 * tid] - W[2 * tid + 1];
    vlds[tid + DIM] = 0.0f;

    // Coalesced tile fill: 64 rows x 32 float4
    const float* src = x + (size_t)tok0 * DIM;
    __builtin_prefetch(src + 64 * DIM, 0, 0);          // gfx1250 global_prefetch_b8
    for (int i = tid; i < 64 * 32; i += 128) {
        int r  = i >> 5;
        int c4 = (i & 31) << 2;
        float4 v = *(const float4*)(src + r * DIM + c4);
        *(float4*)&tile[r * 132 + c4] = v;
    }
    __syncthreads();

    const int wave = tid >> 5;
    const int lane = tid & 31;
    const int lo   = lane & 15;
    const int hi   = lane >> 4;
    const int row  = (wave << 4) + lo;           // A-matrix M-row (token within tile)
    const int vbase = (lo == 0) ? 0 : DIM;       // N!=0 lanes read the zero half
    const float bdiff = bias[0] - bias[1];

    v8f acc = {};
    #pragma unroll
    for (int k = 0; k < 32; ++k) {
        const int col = 4 * k + 2 * hi;          // A 16x4 f32 layout: VGPR0=K{0,2}, VGPR1=K{1,3}
        v2f a  = *(const v2f*)&tile[row * 132 + col];   // ds_load_b64 (col even -> 8B aligned)
        v2f bb = *(const v2f*)&vlds[vbase + col];       // ds_load_b64, zero for N!=0 lanes
        acc = __builtin_amdgcn_wmma_f32_16x16x4_f32(
                  false, a, false, bb, (short)0, acc, false, false);
    }

    // Column N=0 of C: M=0..7 -> lane 0 VGPR 0..7; M=8..15 -> lane 16 VGPR 0..7.
    const int tb = tok0 + (wave << 4);
    unsigned bits = 0u;
    #pragma unroll
    for (int m = 0; m < 16; ++m) {
        float dm = __shfl(acc[m & 7], (m < 8) ? 0 : 16, 32);
        int   t  = tb + m;
        bool  bnd = ((dm + bdiff) > 0.0f) && ((t & (SEQ - 1)) != (SEQ - 1));
        bits |= (bnd ? 1u : 0u) << m;
    }
    if (lane == 0) mask16[tb >> 4] = bits;   // exclusive 16-bit word per wave: no atomics
}

// ----------------------------- Kernel 2: scan -------------------------------
// Events are the first MAXC boundary positions; e_k = min(e_{k-1}+16, t_k+1).
__global__ void tok_scan_kernel(const unsigned* __restrict__ mask16,
                                int* __restrict__ ends,
                                int* __restrict__ nev)
{
    int b = threadIdx.x;
    if (b >= BATCH) return;
    int prev = 0, cnt = 0;
    const unsigned* mw = mask16 + b * (SEQ / 16);
    for (int w = 0; w < SEQ / 16 && cnt < MAXC; ++w) {
        unsigned bits = mw[w];
        int base = w << 4;
        while (bits && cnt < MAXC) {
            int m = __ffs(bits) - 1;
            bits &= bits - 1;
            int t   = base + m;
            int nxt = min(prev + MAXL, t + 1);
            prev = nxt;
            ends[b * MAXC + cnt] = nxt;
            ++cnt;
        }
    }
    nev[b] = cnt;
}

// ----------------------------- Kernel 3: emit -------------------------------
// One block per (batch, chunk). 256 threads cover the 16x128 tile (8 f32 each).
__global__ void tok_emit_kernel(const float* __restrict__ x,
                                const float* __restrict__ pad_vec,
                                const int*   __restrict__ x_ids,
                                const int*   __restrict__ pad_id_p,
                                const int*   __restrict__ ends,
                                const int*   __restrict__ nev,
                                float* __restrict__ out,
                                float* __restrict__ out_ids)
{
    const int blk = blockIdx.x;
    const int b   = blk >> 10;         // / MAXC
    const int k   = blk & (MAXC - 1);
    const int n   = nev[b];

    int start = 0, len = 0;
    if (k < n) {
        int e = ends[b * MAXC + k];
        start = (k > 0) ? ends[b * MAXC + k - 1] : 0;
        len   = e - start;             // 1..16
    }

    const int tid = threadIdx.x;
    const int j   = tid >> 4;          // chunk slot 0..15
    const int d   = (tid & 15) << 3;   // dim offset, 8 floats per thread

    float4 p0 = *(const float4*)(pad_vec + d);
    float4 p1 = *(const float4*)(pad_vec + d + 4);
    if (j < len) {
        const float* xr = x + ((size_t)b * SEQ + start + j) * DIM + d;
        float4 a0 = *(const float4*)xr;
        float4 a1 = *(const float4*)(xr + 4);
        p0.x += a0.x; p0.y += a0.y; p0.z += a0.z; p0.w += a0.w;
        p1.x += a1.x; p1.y += a1.y; p1.z += a1.z; p1.w += a1.w;
    }
    float* o = out + (size_t)blk * (MAXL * DIM) + (size_t)tid * 8;
    *(float4*)o       = p0;
    *(float4*)(o + 4) = p1;

    if ((tid & 15) == 0) {
        float idv = (j < len) ? (float)x_ids[(size_t)b * SEQ + start + j]
                              : (float)pad_id_p[0];
        out_ids[(size_t)blk * MAXL + j] = idv;
    }
}

// ------------------------------- launcher -----------------------------------
extern "C" void kernel_launch(void* const* d_in, const int* in_sizes, int n_in,
                              void* d_out, int out_size, void* d_ws, size_t ws_size,
                              hipStream_t stream)
{
    const float* x      = (const float*)d_in[0];   // (8, 8192, 128)
    const float* padv   = (const float*)d_in[1];   // (128,)
    const int*   ids    = (const int*)  d_in[2];   // (8, 8192)
    const int*   pad_id = (const int*)  d_in[3];   // scalar
    const float* W      = (const float*)d_in[4];   // (128, 2)
    const float* bias   = (const float*)d_in[5];   // (2,)

    float* out     = (float*)d_out;
    float* out_ids = out + (size_t)BATCH * MAXC * MAXL * DIM;

    // workspace: [mask16: 8*512 u32 = 16KB][ends: 8*1024 i32 = 32KB][nev: 8 i32]
    unsigned* mask16 = (unsigned*)d_ws;
    int*      ends   = (int*)((char*)d_ws + 16 * 1024);
    int*      nev    = (int*)((char*)d_ws + 48 * 1024);

    tok_boundary_kernel<<<(BATCH * SEQ) / 64, 128, 0, stream>>>(x, W, bias, mask16);
    tok_scan_kernel<<<1, 32, 0, stream>>>(mask16, ends, nev);
    tok_emit_kernel<<<BATCH * MAXC, 256, 0, stream>>>(x, padv, ids, pad_id,
                                                      ends, nev, out, out_ids);
}